// DCTCompression_25786983645730
// MI455X (gfx1250) — compile-verified
//
#include <hip/hip_runtime.h>

// CDNA5 / gfx1250, wave32. D = A(16x4) x B(4x16) + C via V_WMMA_F32_16X16X4_F32.
typedef float v2f __attribute__((ext_vector_type(2)));
typedef float v8f __attribute__((ext_vector_type(8)));

#define PI_APPROX 3.1415f      // module uses this, not math.pi
#define RS2       0.70710678f  // sqrt(2)/2

// k = ZZ[a*8+u]: standard JPEG zigzag index matrix (derived from the module's
// zigzag(): p = Zstd^T, k = z[a*8+u] = p[u,a] = Zstd[a,u]).
__constant__ int ZZ[64] = {
   0, 1, 5, 6,14,15,27,28,
   2, 4, 7,13,16,26,29,42,
   3, 8,12,17,25,30,41,43,
   9,11,18,24,31,40,44,53,
  10,19,23,32,39,45,52,54,
  20,22,33,38,46,51,55,60,
  21,34,37,47,50,56,59,61,
  35,36,48,49,57,58,62,63};

// Q=50 -> s=100 -> Q_luma == T_LUMA exactly. qvec = 1/table[k].
__constant__ float QTAB[2][64] = {
 {16,11,10,16,24,40,51,61, 12,12,14,19,26,58,60,55, 14,13,16,24,40,57,69,56,
  14,17,22,29,51,87,80,62, 18,22,37,56,68,109,103,77, 24,35,55,64,81,104,113,92,
  49,64,78,87,103,121,120,101, 72,92,95,98,112,100,103,99},
 {17,18,24,47,99,99,99,99, 18,21,26,66,99,99,99,99, 24,26,56,99,99,99,99,99,
  47,66,99,99,99,99,99,99, 99,99,99,99,99,99,99,99, 99,99,99,99,99,99,99,99,
  99,99,99,99,99,99,99,99, 99,99,99,99,99,99,99,99}};

// Block-diagonal 16x16 DCT basis: M16[r,k] = M[r%8][k%8] if same quadrant else 0.
// M[a][y] = cos((a+0.5) * PI_APPROX * y / 8)
__device__ __forceinline__ float m16v(int r, int k) {
  if (((r ^ k) & 8) != 0) return 0.0f;
  return cosf(((float)(r & 7) + 0.5f) * PI_APPROX * (float)(k & 7) * 0.125f);
}

// Wave-private LDS round-trip fence: CDNA5 keeps same-wave LDS ops in order,
// so we only need IR-level memory ordering + a zero-cost scheduling barrier
// (no s_barrier: the 8 waves stay fully decoupled).
__device__ __forceinline__ void wave_lds_fence() {
  __builtin_amdgcn_fence(__ATOMIC_ACQ_REL, "wavefront");
  __builtin_amdgcn_wave_barrier();
}

__global__ __launch_bounds__(256) void dct_zigzag_quant_kernel(
    const float* __restrict__ xin, float* __restrict__ out) {
  // One workgroup = one tile-row of one (b,c) plane:
  //   pixels rows [16*tr, 16*tr+16), all 512 cols -> 32 tiles of 16x16.
  __shared__ float s_out[128 * 64];  // [row = 2*k + hb_local][wb ^ swizzle(row)]
  __shared__ float s_tr[8][320];     // per-wave padded transpose scratch (stride 20)

  const int tid  = threadIdx.x;
  const int lane = tid & 31;
  const int wave = tid >> 5;
  const int hi   = lane >> 4;   // lane >= 16
  const int col  = lane & 15;   // N index / A row index

  const int wg = blockIdx.x;
  const int tr = wg & 31;       // tile row 0..31
  const int bc = wg >> 5;       // b*3 + c
  const int c  = bc % 3;

  // ---- loop-invariant A operand: K-slices of M16 (A layout: M=col, K=v+2*hi+4*j)
  v2f A[4];
#pragma unroll
  for (int j = 0; j < 4; ++j) {
    const int k0 = 4 * j + 2 * hi;
    A[j].x = m16v(col, k0);
    A[j].y = m16v(col, k0 + 1);
  }

  // ---- loop-invariant output constants for this lane
  // After GEMM2, lane holds Y[a_tile = lane%16, u_tile = r + 8*hi], r = 0..7.
  const int alocal = lane & 7;         // DCT row within 8x8 block
  const int qr     = (lane >> 3) & 1;  // vertical block in tile
  const int qc     = hi;               // horizontal block in tile
  const int qsel   = (c == 0) ? 0 : 1;
  int   kk[8];
  float sc[8];
#pragma unroll
  for (int r = 0; r < 8; ++r) {
    const int k = ZZ[alocal * 8 + r];
    kk[r] = k;
    const float nrm = 0.25f * (alocal == 0 ? RS2 : 1.0f) * (r == 0 ? RS2 : 1.0f);
    sc[r] = nrm / QTAB[qsel][k];  // exact f32 divide == reference's 1/table
  }

  const float* xbase = xin + (size_t)bc * (512 * 512) + (size_t)(tr * 16) * 512;
  float* trp = s_tr[wave];

  for (int t = 0; t < 4; ++t) {
    const int tc = wave * 4 + t;  // tile column 0..31
    const float* px = xbase + tc * 16;

    // B operand = X16 (B layout: K = v + 2*hi + 4*j, N = col). 64B-contiguous
    // per half-wave per load.
    v2f Bm[4];
#pragma unroll
    for (int j = 0; j < 4; ++j) {
      const int k0 = 4 * j + 2 * hi;
      Bm[j].x = px[(size_t)k0 * 512 + col];
      Bm[j].y = px[(size_t)(k0 + 1) * 512 + col];
    }

    // GEMM 1: W = M16 * X16  (K=16 as 4 slices of K=4)
    v8f acc = {0.f, 0.f, 0.f, 0.f, 0.f, 0.f, 0.f, 0.f};
#pragma unroll
    for (int j = 0; j < 4; ++j)
      acc = __builtin_amdgcn_wmma_f32_16x16x4_f32(false, A[j], false, Bm[j],
                                                  (short)0, acc, false, false);

    // Transpose through wave-private LDS: W[M = r+8*hi, N = col] stored at
    // P(k=N, n=M) = N*20 + M (stride 20: 16B-aligned b128 stores,
    // conflict-free b32 reads).
    {
      float* dst = trp + col * 20 + 8 * hi;
      float4 lo4 = {acc[0], acc[1], acc[2], acc[3]};
      float4 hi4 = {acc[4], acc[5], acc[6], acc[7]};
      *(float4*)(dst)     = lo4;
      *(float4*)(dst + 4) = hi4;
    }
    wave_lds_fence();  // same-wave LDS ops are in-order; just stop code motion

    // B operand of GEMM 2: Wt[k, n] read from P(k, col)
    v2f B2[4];
#pragma unroll
    for (int j = 0; j < 4; ++j) {
      const int k0 = 4 * j + 2 * hi;
      B2[j].x = trp[k0 * 20 + col];
      B2[j].y = trp[(k0 + 1) * 20 + col];
    }
    wave_lds_fence();  // keep next tile's stores after this tile's reads

    // GEMM 2: Yt = M16 * Wt  ->  lane holds Y[lane%16, r + 8*hi]
    v8f acc2 = {0.f, 0.f, 0.f, 0.f, 0.f, 0.f, 0.f, 0.f};
#pragma unroll
    for (int j = 0; j < 4; ++j)
      acc2 = __builtin_amdgcn_wmma_f32_16x16x4_f32(false, A[j], false, B2[j],
                                                   (short)0, acc2, false, false);

    // Normalize + quantize + zigzag-scatter into staging LDS.
    // Layout: row = 2*k + hb_local (0..127), column = wb XOR-swizzled by row
    // (breaks the 16-way bank conflict of same-wb scatter writes).
    const int wb = 2 * tc + qc;
#pragma unroll
    for (int r = 0; r < 8; ++r) {
      const int row = kk[r] * 2 + qr;
      s_out[row * 64 + (wb ^ ((row & 15) << 2))] = acc2[r] * sc[r];
    }
  }

  __syncthreads();  // the only cross-wave dependency: s_out staging -> drain

  // Drain staging buffer with coalesced float4 stores: each (k, hb) row is a
  // 256B contiguous run in the output plane.
  float* obase = out + (size_t)bc * (64 * 4096) + (size_t)(tr * 2) * 64;
#pragma unroll
  for (int it = 0; it < 8; ++it) {
    const int g    = tid + it * 256;  // float4 index 0..2047
    const int flat = g << 2;
    const int row  = flat >> 6;       // 0..127
    const int wbp  = flat & 63;
    const int wb   = wbp ^ ((row & 15) << 2);  // un-swizzle (multiple of 4)
    const int k    = row >> 1;
    const int hbl  = row & 1;
    const float4 v = *(const float4*)(s_out + flat);
    *(float4*)(obase + (size_t)k * 4096 + hbl * 64 + wb) = v;
  }
}

extern "C" void kernel_launch(void* const* d_in, const int* in_sizes, int n_in,
                              void* d_out, int out_size, void* d_ws, size_t ws_size,
                              hipStream_t stream) {
  (void)n_in; (void)out_size; (void)d_ws; (void)ws_size;
  const float* x = (const float*)d_in[0];
  float* out = (float*)d_out;
  const int batch = in_sizes[0] / (3 * 512 * 512);  // 32 for the reference
  const int grid = batch * 3 * 32;                  // one WG per (b, c, tile-row)
  dct_zigzag_quant_kernel<<<grid, 256, 0, stream>>>(x, out);
}